// BilateralSlicer_19215683682489
// MI455X (gfx1250) — compile-verified
//
#include <hip/hip_runtime.h>
#include <cstdint>

typedef unsigned int v4u __attribute__((ext_vector_type(4)));
typedef int          v8i __attribute__((ext_vector_type(8)));
typedef int          v4i __attribute__((ext_vector_type(4)));
typedef float        v4f __attribute__((ext_vector_type(4)));

namespace {
constexpr int kB = 2, kC = 12, kD = 8, kHG = 16, kWG = 16;
constexpr int kHH = 1080, kWH = 1920;
constexpr int kGridElems    = kC * kD * kHG * kWG;     // 24576 floats = 96 KB per batch
constexpr int kGroupsPerRow = kWH / 4;                 // 480 float4 groups per row
constexpr int kGroupsPerImg = kHH * kGroupsPerRow;     // 518400 groups per batch
constexpr int kBlocksX = 1024;
}

__device__ __forceinline__ int imin(int a, int b) { return a < b ? a : b; }

__global__ __launch_bounds__(256)
void bilateral_slice_kernel(const float* __restrict__ grid,
                            const float* __restrict__ guid,
                            float* __restrict__ out) {
  extern __shared__ float smem[];        // kGridElems floats (96 KB)
  const int b = blockIdx.y;

  // ---- Stage grid[b] (96 KB contiguous) into LDS with the Tensor Data Mover.
  // D# group0: count=1 valid descriptor, lds_addr, 57-bit global addr, type=2.
  // D# group1: data_size=4B (code 2), tensor_dim0 = tile_dim0 = 24576 elems,
  //            tensor_dim1 = tile_dim1 = 1, tensor_dim0_stride = 24576.
  // Groups 2/3 + extra group zero (<=2D tensor). Tracked by TENSORcnt.
  if (threadIdx.x == 0) {
    const uint64_t ga  = (uint64_t)(uintptr_t)(grid + (size_t)b * kGridElems);
    const uint32_t lds = (uint32_t)(uintptr_t)smem;  // LDS byte offset = low 32 bits of flat addr
    v4u g0 = { 1u,                                    // [1:0] count=1, rest 0
               lds,                                   // [63:32] lds_addr
               (uint32_t)ga,                          // global_addr[31:0]
               (uint32_t)((ga >> 32) & 0x01ffffffu) | (2u << 30) }; // addr[56:32] | type=2
    v8i g1 = { (int)(2u << 16),                       // [17:16] data_size = 4 bytes
               (int)((uint32_t)kGridElems << 16),     // [79:48] tensor_dim0 (low 16 here)
               (int)(1u << 16),                       // [111:80] tensor_dim1 = 1
               (int)((uint32_t)kGridElems << 16),     // [127:112] tile_dim0 = 24576
               1,                                     // [143:128] tile_dim1 = 1
               kGridElems,                            // [207:160] tensor_dim0_stride (low 32)
               0, 0 };
    v4i g2 = {0, 0, 0, 0};
    v4i g3 = {0, 0, 0, 0};
    v8i g4 = {0, 0, 0, 0, 0, 0, 0, 0};
    __builtin_amdgcn_tensor_load_to_lds(g0, g1, g2, g3, g4, /*cpol=*/0);
    __builtin_amdgcn_s_wait_tensorcnt(0);
  }
  __syncthreads();

  const float sy = (float)kHG / (float)kHH;   // 16/1080
  const float sx = (float)kWG / (float)kWH;   // 16/1920

  for (int g = blockIdx.x * 256 + threadIdx.x; g < kGroupsPerImg;
       g += kBlocksX * 256) {
    const int y  = g / kGroupsPerRow;
    const int x4 = (g - y * kGroupsPerRow) * 4;

    // y-interp weights: shared by the 4 pixels of this group (same row).
    float fy = fmaxf(fmaf((float)y + 0.5f, sy, -0.5f), 0.0f);
    int   y0  = imin((int)fy, kHG - 1);
    int   y1  = imin(y0 + 1, kHG - 1);
    float wy1 = fy - (float)y0, wy0 = 1.0f - wy1;

    // 4 guidance values in one non-temporal b128 load (read-once stream).
    const v4f gd = __builtin_nontemporal_load(
        (const v4f*)(guid + ((size_t)b * kHH + y) * kWH + x4));

    int   off[4][8];
    float wgt[4][8];
#pragma unroll
    for (int p = 0; p < 4; ++p) {
      const int x = x4 + p;
      float fx = fmaxf(fmaf((float)x + 0.5f, sx, -0.5f), 0.0f);
      int   x0  = imin((int)fx, kWG - 1);
      int   x1  = imin(x0 + 1, kWG - 1);
      float wx1 = fx - (float)x0, wx0 = 1.0f - wx1;

      float fz  = gd[p] * (float)(kD - 1);
      float fzf = fminf(fmaxf(floorf(fz), 0.0f), (float)(kD - 1));
      int   z0  = (int)fzf;
      int   z1  = imin(z0 + 1, kD - 1);
      float wz1 = fz - fzf, wz0 = 1.0f - wz1;

      const int r00 = z0 * (kHG * kWG) + y0 * kWG;
      const int r01 = z0 * (kHG * kWG) + y1 * kWG;
      const int r10 = z1 * (kHG * kWG) + y0 * kWG;
      const int r11 = z1 * (kHG * kWG) + y1 * kWG;
      off[p][0] = r00 + x0;  wgt[p][0] = wz0 * wy0 * wx0;
      off[p][1] = r00 + x1;  wgt[p][1] = wz0 * wy0 * wx1;
      off[p][2] = r01 + x0;  wgt[p][2] = wz0 * wy1 * wx0;
      off[p][3] = r01 + x1;  wgt[p][3] = wz0 * wy1 * wx1;
      off[p][4] = r10 + x0;  wgt[p][4] = wz1 * wy0 * wx0;
      off[p][5] = r10 + x1;  wgt[p][5] = wz1 * wy0 * wx1;
      off[p][6] = r11 + x0;  wgt[p][6] = wz1 * wy1 * wx0;
      off[p][7] = r11 + x1;  wgt[p][7] = wz1 * wy1 * wx1;
    }

    // 12 channels: 8 LDS broadcast loads + 8 FMA per pixel, one b128 NT store
    // per channel (write-once 199 MB stream -> keep out of caches).
    float* op = out + (((size_t)b * kC) * kHH + y) * kWH + x4;
#pragma unroll
    for (int c = 0; c < kC; ++c) {
      const float* gc = smem + c * (kD * kHG * kWG);
      v4f r;
#pragma unroll
      for (int p = 0; p < 4; ++p) {
        float acc = 0.0f;
#pragma unroll
        for (int k = 0; k < 8; ++k)
          acc = fmaf(wgt[p][k], gc[off[p][k]], acc);
        r[p] = acc;
      }
      __builtin_nontemporal_store(r, (v4f*)(op + (size_t)c * (kHH * kWH)));
    }
  }
}

extern "C" void kernel_launch(void* const* d_in, const int* in_sizes, int n_in,
                              void* d_out, int out_size, void* d_ws, size_t ws_size,
                              hipStream_t stream) {
  const float* grid = (const float*)d_in[0];   // (2,12,8,16,16) f32
  const float* guid = (const float*)d_in[1];   // (2,1,1080,1920) f32
  float* out = (float*)d_out;                  // (2,12,1080,1920) f32

  dim3 gdim(kBlocksX, kB);
  size_t shmem = (size_t)kGridElems * sizeof(float);   // 96 KB dynamic LDS
  bilateral_slice_kernel<<<gdim, 256, shmem, stream>>>(grid, guid, out);
}